// Criterion_63960652972293
// MI455X (gfx1250) — compile-verified
//
#include <hip/hip_runtime.h>
#include <cfloat>
#include <cstdint>

// ---------------- problem constants (from reference setup) ----------------
#define C_CH 19
#define H_IMG 512
#define W_IMG 1024
#define HW (H_IMG * W_IMG)          // 524288
#define B_SZ 8
#define P_TOT (B_SZ * HW)           // 4194304 pixels
#define TILE 256
#define NTILES (P_TOT / TILE)       // 16384 (HW % TILE == 0 -> no batch crossing)
#define OHEM_GRID 2048
#define TILES_PER_BLK (NTILES / OHEM_GRID)   // 8, exact
#define NBINS 512
#define OHEM_THRESH 0.22314355f     // -log(0.8)
#define HH 128
#define WH 256
#define HM_TOT (B_SZ * HH * WH)     // 262144
#define N_BOX 64
#define BN_TOT (B_SZ * N_BOX)       // 512

// ws float layout:
// [0] sum_hard  [1] cnt_hard  [2] cnt_valid
// [3] pos_loss  [4] neg_loss  [5] n_pos  [6] l1_sum  [7] pad
// [8 .. 8+NBINS)           hist counts (sub-threshold ce)
// [8+NBINS .. 8+2*NBINS)   hist sums
#define WS_SCALARS 8
#define WS_TOTAL (WS_SCALARS + 2 * NBINS)

// ---------------- CDNA5 async global->LDS path (probe-confirmed present) ----------------
#if defined(__AMDGCN__) && __has_builtin(__builtin_amdgcn_global_load_async_to_lds_b128) && __has_builtin(__builtin_amdgcn_s_wait_asynccnt)
#define ASYNC_OK 1
#else
#define ASYNC_OK 0
#endif

#if ASYNC_OK
typedef int v4i_t __attribute__((ext_vector_type(4)));
typedef __attribute__((address_space(1))) v4i_t gv4i_t;   // global
typedef __attribute__((address_space(3))) v4i_t lv4i_t;   // LDS
#endif

__device__ __forceinline__ void stage_b128(const float* __restrict__ g, float* __restrict__ l) {
#if ASYNC_OK
    __builtin_amdgcn_global_load_async_to_lds_b128((gv4i_t*)g, (lv4i_t*)l, 0, 0);
#else
    *(float4*)l = *(const float4*)g;
#endif
}

// Wait for all transfers of the PREVIOUS tile: async loads complete in order,
// so ASYNCcnt <= (issues just made for the new tile) drains everything older.
// Waves 0-2 stage 3 channels (6 issues), waves 3-7 stage 2 (4 issues).
__device__ __forceinline__ void stage_wait_pipe(int wave, bool pending) {
#if ASYNC_OK
    if (pending) {
        if (wave < 3) __builtin_amdgcn_s_wait_asynccnt(6);
        else          __builtin_amdgcn_s_wait_asynccnt(4);
    } else {
        __builtin_amdgcn_s_wait_asynccnt(0);
    }
#else
    (void)wave; (void)pending;
#endif
}

// Issue one 256-pixel tile (19 channel rows of 1 KB) into LDS buffer sb.
__device__ __forceinline__ void issue_tile(const float* __restrict__ seg,
                                           float* __restrict__ sb,
                                           int tile, int wave, int lane) {
    const int pbase = tile * TILE;
    const int b  = pbase / HW;
    const int hw = pbase - b * HW;
    const size_t segbase = (size_t)b * C_CH * HW + hw;
    for (int c = wave; c < C_CH; c += 8) {
        const float* gsrc = seg + segbase + (size_t)c * HW;
        float* ldst = sb + c * TILE;
        stage_b128(gsrc + lane * 4,       ldst + lane * 4);
        stage_b128(gsrc + 128 + lane * 4, ldst + 128 + lane * 4);
    }
}

// ---------------- kernel 0: zero the workspace accumulators ----------------
__global__ __launch_bounds__(256) void k_init(float* __restrict__ ws) {
    int i = blockIdx.x * blockDim.x + threadIdx.x;
    if (i < WS_TOTAL) ws[i] = 0.0f;
}

// ---------------- kernel 1: OHEM CE pass (bandwidth-dominant, 2-stage async pipeline) ----
__global__ __launch_bounds__(256) void k_ohem(const float* __restrict__ seg,
                                              const int* __restrict__ masks,
                                              float* __restrict__ ws) {
    __shared__ float smem[2][C_CH * TILE];  // double-buffered 19 KB tile stage
    __shared__ float h_cnt[NBINS];
    __shared__ float h_sum[NBINS];
    __shared__ float blk[3];                // sum_hard, cnt_hard, cnt_valid

    const int tid  = threadIdx.x;
    const int wave = tid >> 5;
    const int lane = tid & 31;

    for (int i = tid; i < NBINS; i += 256) { h_cnt[i] = 0.0f; h_sum[i] = 0.0f; }
    if (tid < 3) blk[tid] = 0.0f;
    __syncthreads();

    float s_hard = 0.0f, c_hard = 0.0f, c_valid = 0.0f;

    // ---- prologue: stage tile 0 into buffer 0 ----
    issue_tile(seg, smem[0], blockIdx.x, wave, lane);

    for (int i = 0; i < TILES_PER_BLK; ++i) {
        const int tile = blockIdx.x + i * OHEM_GRID;
        const bool have_next = (i + 1 < TILES_PER_BLK);

        // ---- stage tile i+1 into the other buffer while tile i is consumed ----
        if (have_next)
            issue_tile(seg, smem[(i + 1) & 1], blockIdx.x + (i + 1) * OHEM_GRID, wave, lane);
        stage_wait_pipe(wave, have_next);   // drains tile i's transfers
        __syncthreads();                    // all waves' data for tile i visible

        // GL2 pre-warm for tile i+2 (global_prefetch_b8)
        if (i + 2 < TILES_PER_BLK) {
            int pb2 = (blockIdx.x + (i + 2) * OHEM_GRID) * TILE;
            int b2  = pb2 / HW;
            int hw2 = pb2 - b2 * HW;
            __builtin_prefetch(seg + (size_t)b2 * C_CH * HW + hw2 + (size_t)wave * HW, 0, 3);
        }

        const float* sb = smem[i & 1];

        // ---- per-pixel logsumexp + CE ----
        const int p = tile * TILE + tid;
        const int t = masks[p];
        const bool valid = (t != 255);

        float x[C_CH];
        #pragma unroll
        for (int c = 0; c < C_CH; ++c) x[c] = sb[c * TILE + tid];

        float m = x[0];
        #pragma unroll
        for (int c = 1; c < C_CH; ++c) m = fmaxf(m, x[c]);
        float sum = 0.0f;
        #pragma unroll
        for (int c = 0; c < C_CH; ++c) sum += __expf(x[c] - m);

        const int tc  = valid ? t : 0;
        const float xt = sb[tc * TILE + tid];            // one dynamic LDS read
        const float ce = valid ? (m + __logf(sum) - xt) : 0.0f;

        if (valid) {
            c_valid += 1.0f;
            if (ce > OHEM_THRESH) {
                s_hard += ce;
                c_hard += 1.0f;
            } else {
                int bin = (int)(ce * ((float)NBINS / OHEM_THRESH));
                bin = bin < 0 ? 0 : (bin > NBINS - 1 ? NBINS - 1 : bin);
                atomicAdd(&h_cnt[bin], 1.0f);
                atomicAdd(&h_sum[bin], ce);
            }
        }
        __syncthreads();   // all waves done reading buffer i&1 -> reusable at i+2
    }

    // ---- wave reduce, then block, then global ----
    #pragma unroll
    for (int off = 16; off; off >>= 1) {
        s_hard  += __shfl_down(s_hard,  off, 32);
        c_hard  += __shfl_down(c_hard,  off, 32);
        c_valid += __shfl_down(c_valid, off, 32);
    }
    if (lane == 0) {
        atomicAdd(&blk[0], s_hard);
        atomicAdd(&blk[1], c_hard);
        atomicAdd(&blk[2], c_valid);
    }
    __syncthreads();
    if (tid == 0) {
        atomicAdd(&ws[0], blk[0]);
        atomicAdd(&ws[1], blk[1]);
        atomicAdd(&ws[2], blk[2]);
    }
    for (int i = tid; i < NBINS; i += 256) {
        if (h_cnt[i] != 0.0f) {
            atomicAdd(&ws[WS_SCALARS + i],         h_cnt[i]);
            atomicAdd(&ws[WS_SCALARS + NBINS + i], h_sum[i]);
        }
    }
}

// ---------------- kernel 2: focal loss partial sums ----------------
__global__ __launch_bounds__(256) void k_focal(const float* __restrict__ pred,
                                               const float* __restrict__ tgt,
                                               float* __restrict__ ws) {
    __shared__ float sblk[3];
    const int tid = threadIdx.x;
    const int lane = tid & 31;
    if (tid < 3) sblk[tid] = 0.0f;
    __syncthreads();

    float pl = 0.0f, nl = 0.0f, np = 0.0f;
    for (int i = blockIdx.x * blockDim.x + tid; i < HM_TOT; i += gridDim.x * blockDim.x) {
        const float t = tgt[i];
        const float p = fmaxf(pred[i], FLT_MIN);      // finfo(f32).tiny
        const float pos = (t == 1.0f) ? 1.0f : 0.0f;
        const float neg = (t < 1.0f) ? 1.0f : 0.0f;
        const float om  = 1.0f - t;
        const float negw = om * om * om * om;
        const float omp  = 1.0f - p;
        pl += __logf(p) * omp * omp * pos;
        nl += __logf(1.0f - p + FLT_MIN) * p * p * negw * neg;
        np += pos;
    }
    #pragma unroll
    for (int off = 16; off; off >>= 1) {
        pl += __shfl_down(pl, off, 32);
        nl += __shfl_down(nl, off, 32);
        np += __shfl_down(np, off, 32);
    }
    if (lane == 0) {
        atomicAdd(&sblk[0], pl);
        atomicAdd(&sblk[1], nl);
        atomicAdd(&sblk[2], np);
    }
    __syncthreads();
    if (tid == 0) {
        atomicAdd(&ws[3], sblk[0]);
        atomicAdd(&ws[4], sblk[1]);
        atomicAdd(&ws[5], sblk[2]);
    }
}

// ---------------- kernel 3: L1 regression over 512 gathered points ----------------
__global__ __launch_bounds__(512) void k_reg(const float* __restrict__ wh,
                                             const float* __restrict__ bb,
                                             const int* __restrict__ ct,
                                             float* __restrict__ ws) {
    __shared__ float wsum[16];
    const int i = threadIdx.x;           // 0..511 == b*64 + n
    const int wave = i >> 5, lane = i & 31;

    float l1 = 0.0f;
    if (i < BN_TOT) {
        const int b = i >> 6;
        const int x = ct[i * 2 + 0];
        const int y = ct[i * 2 + 1];
        const float* whb = wh + (size_t)b * 2 * HH * WH;
        const float v0 = whb[y * WH + x];             // channel 0
        const float v1 = whb[HH * WH + y * WH + x];   // channel 1
        const float w = bb[i * 4 + 2] - bb[i * 4 + 0];
        const float h = bb[i * 4 + 3] - bb[i * 4 + 1];
        l1 = fabsf(v0 - w) + fabsf(v1 - h);
    }
    #pragma unroll
    for (int off = 16; off; off >>= 1) l1 += __shfl_down(l1, off, 32);
    if (lane == 0) wsum[wave] = l1;
    __syncthreads();
    if (wave == 0) {
        float v = (lane < 16) ? wsum[lane] : 0.0f;
        #pragma unroll
        for (int off = 8; off; off >>= 1) v += __shfl_down(v, off, 32);
        if (lane == 0) atomicAdd(&ws[6], v);
    }
}

// ---------------- kernel 4: scalar combine ----------------
__global__ void k_final(const float* __restrict__ ws, float* __restrict__ out) {
    if (threadIdx.x != 0 || blockIdx.x != 0) return;

    const float sum_hard = ws[0], cnt_hard = ws[1], cnt_valid = ws[2];
    const float n_min = floorf(cnt_valid * (1.0f / 16.0f));

    float k, top;
    if (cnt_hard >= n_min) {                 // common case: top-k == all hard examples
        k = cnt_hard;
        top = sum_hard;
    } else {                                 // fill up to n_min from histogram (desc)
        k = n_min;
        top = sum_hard;
        float need = n_min - cnt_hard;
        for (int bin = NBINS - 1; bin >= 0 && need > 0.0f; --bin) {
            const float c = ws[WS_SCALARS + bin];
            const float s = ws[WS_SCALARS + NBINS + bin];
            if (c <= need) { top += s; need -= c; }
            else           { top += s * (need / c); need = 0.0f; }
        }
    }
    const float cls = (k > 0.0f) ? (top / k) : 0.0f;   // * CLASSIFICATION_WEIGHT(1.0)

    const float pl = ws[3], nl = ws[4], np = ws[5];
    const float centerness = (np == 0.0f) ? -nl : -(pl + nl) / fmaxf(np, 1.0f);

    const float bbox = (ws[6] * 0.7f) / ((float)BN_TOT + 1e-7f) * 0.1f;
    const float loc  = (centerness + bbox) * 1.0f;     // LOCALIZATION_WEIGHT

    out[0] = cls;
    out[1] = loc;
    out[2] = centerness;
    out[3] = bbox;
}

// ---------------- host-side launch ----------------
extern "C" void kernel_launch(void* const* d_in, const int* in_sizes, int n_in,
                              void* d_out, int out_size, void* d_ws, size_t ws_size,
                              hipStream_t stream) {
    const float* seg   = (const float*)d_in[0];  // (8,19,512,1024)
    const int*   masks = (const int*)  d_in[1];  // (8,512,1024)
    const float* hp    = (const float*)d_in[2];  // (8,1,128,256)
    const float* hm    = (const float*)d_in[3];  // (8,1,128,256)
    const float* wh    = (const float*)d_in[4];  // (8,2,128,256)
    const float* bb    = (const float*)d_in[5];  // (8,64,4)
    const int*   ct    = (const int*)  d_in[7];  // (8,64,2)  (labels d_in[6] unused by ref)
    float* out = (float*)d_out;
    float* ws  = (float*)d_ws;

    (void)in_sizes; (void)n_in; (void)out_size; (void)ws_size;

    k_init <<<(WS_TOTAL + 255) / 256, 256, 0, stream>>>(ws);
    k_ohem <<<OHEM_GRID, 256, 0, stream>>>(seg, masks, ws);   // 8 tiles/block, pipelined
    k_focal<<<256, 256, 0, stream>>>(hp, hm, ws);
    k_reg  <<<1, 512, 0, stream>>>(wh, bb, ct, ws);
    k_final<<<1, 32, 0, stream>>>(ws, out);
}